// SpatioTemporalOutageModel_55783035240790
// MI455X (gfx1250) — compile-verified
//
#include <hip/hip_runtime.h>

// ---------------------------------------------------------------------------
// SpatioTemporal GCN + LSTM for MI455X (gfx1250, wave32, WMMA f32 16x16x4,
// async global->LDS double-buffered embeds stream in the persistent LSTM)
// ---------------------------------------------------------------------------

#define FEAT 14
#define DD   64
#define HHID 128
#define NGATE 512

typedef float v2f __attribute__((ext_vector_type(2)));
typedef float v8f __attribute__((ext_vector_type(8)));
typedef int   v4i_vs __attribute__((vector_size(16)));   // matches builtin param

#ifndef __has_builtin
#define __has_builtin(x) 0
#endif

#if __has_builtin(__builtin_amdgcn_global_load_async_to_lds_b128) && \
    __has_builtin(__builtin_amdgcn_s_wait_asynccnt)
#define EMB_ASYNC 1
#else
#define EMB_ASYNC 0
#endif

// builtin expects v4i pointers in explicit address spaces (per hipcc diag)
#define AS1PTR(p) ((__attribute__((address_space(1))) v4i_vs*)(unsigned long long)(p))
#define AS3PTR(p) ((__attribute__((address_space(3))) v4i_vs*)(unsigned int)(unsigned long long)(p))

static __device__ __forceinline__ v8f wmma4(v2f a, v2f b, v8f c) {
  // D = A(16x4 f32) * B(4x16 f32) + C(16x16 f32)
  return __builtin_amdgcn_wmma_f32_16x16x4_f32(
      /*neg_a=*/false, a, /*neg_b=*/false, b,
      /*c_mod=*/(short)0, c, /*reuse_a=*/false, /*reuse_b=*/false);
}

static __device__ __forceinline__ float sigmoidf_(float x) {
  return 1.0f / (1.0f + __expf(-x));
}

// ---------------------------------------------------------------------------
// CSR construction (deterministic): count -> scan -> scatter -> bucket sort
// ---------------------------------------------------------------------------
__global__ void k_count(const int* __restrict__ dst, int* __restrict__ cnt,
                        int E, int N) {
  int i = blockIdx.x * blockDim.x + threadIdx.x;
  if (i < E + N) {
    int d = (i < E) ? dst[i] : (i - E);   // ids >= E are self-loops
    atomicAdd(&cnt[d], 1);                // integer: order-independent
  }
}

__global__ void k_scan(const int* __restrict__ cnt, int* __restrict__ ptr,
                       int* __restrict__ cur, int N) {
  if (blockIdx.x == 0 && threadIdx.x == 0) {
    int acc = 0;
    for (int n = 0; n < N; ++n) { ptr[n] = acc; cur[n] = acc; acc += cnt[n]; }
    ptr[N] = acc;
  }
}

__global__ void k_scatter(const int* __restrict__ dst, int* __restrict__ cur,
                          int* __restrict__ eid, int E, int N) {
  int i = blockIdx.x * blockDim.x + threadIdx.x;
  if (i < E + N) {
    int d = (i < E) ? dst[i] : (i - E);
    int pos = atomicAdd(&cur[d], 1);
    eid[pos] = i;
  }
}

// sort each bucket by edge id (fixed accumulation order) + degree + rsqrt
__global__ void k_sort_deg(int* __restrict__ eid, const int* __restrict__ ptr,
                           const float* __restrict__ ew, float* __restrict__ dis,
                           int E, int N) {
  int n = blockIdx.x * blockDim.x + threadIdx.x;
  if (n >= N) return;
  int b = ptr[n], e2 = ptr[n + 1];
  for (int i = b + 1; i < e2; ++i) {          // insertion sort (avg ~7 elems)
    int key = eid[i]; int j = i - 1;
    while (j >= b && eid[j] > key) { eid[j + 1] = eid[j]; --j; }
    eid[j + 1] = key;
  }
  float deg = 0.f;
  for (int p = b; p < e2; ++p) {
    int e = eid[p];
    deg += (e < E) ? ew[e] : 1.0f;            // self-loop weight 1
  }
  dis[n] = (deg > 0.f) ? rsqrtf(deg) : 0.f;
}

__global__ void k_norm(const int* __restrict__ src, const int* __restrict__ dst,
                       const float* __restrict__ ew, const float* __restrict__ dis,
                       float* __restrict__ normw, int E, int N) {
  int i = blockIdx.x * blockDim.x + threadIdx.x;
  if (i < E + N) {
    if (i < E) normw[i] = dis[src[i]] * ew[i] * dis[dst[i]];
    else       { int n = i - E; normw[i] = dis[n] * dis[n]; }
  }
}

// ---------------------------------------------------------------------------
// GCN1 transform: [T*N,14] @ [14,64]  (K=14: memory bound, plain FMAs)
// ---------------------------------------------------------------------------
__global__ void k_hw1(const float* __restrict__ x, const float* __restrict__ W1,
                      float* __restrict__ out, long rows) {
  long idx = (long)blockIdx.x * blockDim.x + threadIdx.x;
  if (idx >= rows * 64) return;
  long row = idx >> 6; int d = (int)(idx & 63);
  const float* xr = x + row * FEAT;
  float acc = 0.f;
#pragma unroll
  for (int f = 0; f < FEAT; ++f) acc += xr[f] * W1[f * 64 + d];
  out[idx] = acc;
}

// ---------------------------------------------------------------------------
// Edge aggregation via CSR (no float atomics, deterministic)
//   out[t,n,d] = sum_{e in in(n)} norm[e] * hw[t, src(e), d] (+bias +county, relu)
// block = 256 threads = 4 nodes x 64 channels
// ---------------------------------------------------------------------------
__global__ void k_agg(const float* __restrict__ hw, const int* __restrict__ ptr,
                      const int* __restrict__ eid, const int* __restrict__ src,
                      const float* __restrict__ normw, const float* __restrict__ bias,
                      const float* __restrict__ county, int use_county, int do_relu,
                      float* __restrict__ out, int N, int E) {
  int t = blockIdx.y;
  int n = blockIdx.x * 4 + (threadIdx.x >> 6);
  int d = threadIdx.x & 63;
  if (n >= N) return;
  int b = ptr[n], e2 = ptr[n + 1];
  const float* base = hw + (long)t * N * 64;
  float acc = 0.f;
  for (int p = b; p < e2; ++p) {
    int e = eid[p];
    int s = (e < E) ? src[e] : (e - E);
    acc += normw[e] * base[(long)s * 64 + d];
  }
  float v = acc + bias[d];
  if (use_county) v += county[(long)n * 64 + d];
  if (do_relu) v = fmaxf(v, 0.f);
  out[(long)t * N * 64 + (long)n * 64 + d] = v;
}

// ---------------------------------------------------------------------------
// GCN2 transform with WMMA f32 16x16x4: [T*N,64] @ [64,64]
// 8 waves/block, each wave: one 16-row tile x 4 col-tiles, K=16 wmma each.
// W2^T in LDS, row stride 68 (68 mod 64 == 4 -> conflict-free float2 reads).
// ---------------------------------------------------------------------------
__global__ void k_hw2(const float* __restrict__ hin, const float* __restrict__ W2,
                      float* __restrict__ out, int ntiles) {
  __shared__ float w2t[64 * 68];
  for (int idx = threadIdx.x; idx < 64 * 64; idx += 256) {
    int k = idx >> 6, n2 = idx & 63;
    w2t[n2 * 68 + k] = W2[idx];               // store transposed: w2t[n][k]
  }
  __syncthreads();
  int wv = threadIdx.x >> 5, lane = threadIdx.x & 31;
  int tile = blockIdx.x * 8 + wv;
  if (tile >= ntiles) return;
  long rowbase = (long)tile * 16;
  int m = lane & 15;                          // A: lane->row M; B/D: lane->col N
  int dl = (lane >> 4) * 2;                   // K offset for upper half-wave
  const float* arow = hin + (rowbase + m) * 64 + dl;
  v8f acc[4] = {};
#pragma unroll
  for (int kk = 0; kk < 16; ++kk) {
    v2f a = *(const v2f*)(arow + 4 * kk);
#pragma unroll
    for (int g = 0; g < 4; ++g) {
      v2f b = *(const v2f*)&w2t[(g * 16 + m) * 68 + 4 * kk + dl];
      acc[g] = wmma4(a, b, acc[g]);
    }
  }
#pragma unroll
  for (int g = 0; g < 4; ++g)
#pragma unroll
    for (int r = 0; r < 8; ++r) {
      int M = r + ((lane >> 4) << 3);         // C/D layout: lanes 16-31 -> M+8
      out[(rowbase + M) * 64 + g * 16 + m] = acc[g][r];
    }
}

// ---------------------------------------------------------------------------
// Embeds-tile loader for the LSTM: one B128 async global->LDS op per wave
// (wave wv covers rows 2wv, 2wv+1; lane -> 16B chunk). Fallback: float4 copy.
// ---------------------------------------------------------------------------
static __device__ __forceinline__ void emb_issue(float* dstbuf,
                                                 const float* __restrict__ esrc,
                                                 int nbase, int N, int wv, int lane) {
  int row  = 2 * wv + (lane >> 4);
  int c4   = (lane & 15) * 4;
  int node = nbase + row; if (node >= N) node = N - 1;     // clamp (store guarded)
  const float* g = esrc + (long)node * 64 + c4;
  float* l = dstbuf + row * 68 + c4;
#if EMB_ASYNC
  __builtin_amdgcn_global_load_async_to_lds_b128(AS1PTR(g), AS3PTR(l),
                                                 /*offset=*/0, /*cpol=*/0);
#else
  *(float4*)l = *(const float4*)g;
#endif
}

static __device__ __forceinline__ void emb_wait() {
#if EMB_ASYNC
  __builtin_amdgcn_s_wait_asynccnt(0);
#endif
}

// ---------------------------------------------------------------------------
// Persistent LSTM: one block per 16-node tile, loops all 336 timesteps.
//  - W_hh (512x128) resident in LDS, row stride 132 (conflict-free float2)
//  - each wave owns 16 hidden units => 4 gate col-tiles (i,f,g,o)
//  - embeds tiles double-buffered, streamed via async global->LDS (ASYNCcnt)
//  - cell state c in accumulator-layout registers; h exchanged via LDS
//  - 192 v_wmma_f32_16x16x4_f32 per wave per step
// LDS: 512*132 + 16*132 + 2*16*68 = 71872 floats = 287488 bytes (< 320KB WGP)
// ---------------------------------------------------------------------------
__global__ void __launch_bounds__(256, 1)
k_lstm(const float* __restrict__ emb, const float* __restrict__ W_ih,
       const float* __restrict__ W_hh, const float* __restrict__ b_ih,
       const float* __restrict__ b_hh, float* __restrict__ hfinal, int T, int N) {
  extern __shared__ float smem[];
  float* whh   = smem;                    // [512][132]
  float* hbuf  = smem + 512 * 132;        // [16][132]
  float* ebuf0 = hbuf + 16 * 132;         // [16][68]
  float* ebuf1 = ebuf0 + 16 * 68;         // [16][68]
  int tid = threadIdx.x;
  int lane = tid & 31, wv = tid >> 5;
  int m = lane & 15, dl = (lane >> 4) * 2;
  int nbase = blockIdx.x * 16;

  // prefetch embeds tile for t=0 (overlaps with the W_hh LDS fill below)
  emb_issue(ebuf0, emb, nbase, N, wv, lane);

  for (int idx = tid; idx < 512 * 128; idx += 256) {
    int n2 = idx >> 7, k = idx & 127;
    whh[n2 * 132 + k] = W_hh[idx];
  }
  for (int idx = tid; idx < 16 * 132; idx += 256) hbuf[idx] = 0.f;

  int cb[4]; float biasv[4]; int hwoff[4];
  v2f wih[16][4];
#pragma unroll
  for (int g = 0; g < 4; ++g) {
    cb[g] = g * 128 + wv * 16;            // gate col-tile base (i,f,g,o blocks)
    int col = cb[g] + m;
    biasv[g] = b_ih[col] + b_hh[col];
    hwoff[g] = col * 132 + dl;
#pragma unroll
    for (int kk = 0; kk < 16; ++kk)
      wih[kk][g] = *(const v2f*)(W_ih + col * 64 + 4 * kk + dl);
  }
  float cst[8];
#pragma unroll
  for (int r = 0; r < 8; ++r) cst[r] = 0.f;
  const int eoff = m * 68 + dl;
  const int hoff = m * 132 + dl;
  const int ucol = wv * 16 + m;

  for (int t = 0; t < T; ++t) {
    float* cbuf = (t & 1) ? ebuf1 : ebuf0;
    float* nbuf = (t & 1) ? ebuf0 : ebuf1;
    emb_wait();        // this wave's pending async tile (into cbuf) landed
    __syncthreads();   // B1: all tiles + prev-step h writes visible everywhere
    if (t + 1 < T)     // prefetch next step's tile; overlaps the WMMA chain
      emb_issue(nbuf, emb + (long)(t + 1) * N * 64, nbase, N, wv, lane);

    v8f acc[4];
#pragma unroll
    for (int g = 0; g < 4; ++g) {
      v8f z;
#pragma unroll
      for (int r = 0; r < 8; ++r) z[r] = biasv[g];
      acc[g] = z;
    }
    // gates += emb_tile(16x64) @ W_ih_slice^T   (K = 64)
#pragma unroll
    for (int kk = 0; kk < 16; ++kk) {
      v2f a = *(const v2f*)&cbuf[eoff + 4 * kk];
#pragma unroll
      for (int g = 0; g < 4; ++g) acc[g] = wmma4(a, wih[kk][g], acc[g]);
    }
    // gates += h_tile(16x128) @ W_hh_slice^T    (K = 128)
#pragma unroll
    for (int kk = 0; kk < 32; ++kk) {
      v2f a = *(const v2f*)&hbuf[hoff + 4 * kk];
#pragma unroll
      for (int g = 0; g < 4; ++g) {
        v2f b = *(const v2f*)&whh[hwoff[g] + 4 * kk];
        acc[g] = wmma4(a, b, acc[g]);
      }
    }
    float hv[8];
#pragma unroll
    for (int r = 0; r < 8; ++r) {
      float ig = sigmoidf_(acc[0][r]);
      float fg = sigmoidf_(acc[1][r]);
      float gg = tanhf(acc[2][r]);
      float og = sigmoidf_(acc[3][r]);
      cst[r] = fg * cst[r] + ig * gg;
      hv[r]  = og * tanhf(cst[r]);
    }
    __syncthreads();   // B2: all h reads of this step complete before overwrite
#pragma unroll
    for (int r = 0; r < 8; ++r) {
      int M = r + ((lane >> 4) << 3);
      hbuf[M * 132 + ucol] = hv[r];
    }
  }
  __syncthreads();
  for (int idx = tid; idx < 16 * 128; idx += 256) {
    int r = idx >> 7, u = idx & 127;
    int node = nbase + r;
    if (node < N) hfinal[(long)node * 128 + u] = hbuf[r * 132 + u];
  }
}

// ---------------------------------------------------------------------------
// Head MLP: relu(h @ Wm1 + bm1) @ Wm2 + bm2   (tiny; one block per node)
// ---------------------------------------------------------------------------
__global__ void k_head(const float* __restrict__ hfinal, const float* __restrict__ Wm1,
                       const float* __restrict__ bm1, const float* __restrict__ Wm2,
                       const float* __restrict__ bm2, float* __restrict__ out, int N) {
  __shared__ float red[64];
  int n = blockIdx.x; int j = threadIdx.x;
  const float* hr = hfinal + (long)n * 128;
  float acc = bm1[j];
#pragma unroll 8
  for (int k = 0; k < 128; ++k) acc += hr[k] * Wm1[k * 64 + j];
  red[j] = fmaxf(acc, 0.f) * Wm2[j];
  __syncthreads();
  if (j == 0) {
    float s = 0.f;
    for (int i = 0; i < 64; ++i) s += red[i];   // fixed order: deterministic
    out[n] = s + bm2[0];
  }
}

// ---------------------------------------------------------------------------
extern "C" void kernel_launch(void* const* d_in, const int* in_sizes, int n_in,
                              void* d_out, int out_size, void* d_ws, size_t ws_size,
                              hipStream_t stream) {
  const float* x      = (const float*)d_in[0];
  const int*   ei     = (const int*)  d_in[1];
  const float* ew     = (const float*)d_in[2];
  const float* W1     = (const float*)d_in[3];
  const float* b1     = (const float*)d_in[4];
  const float* W2     = (const float*)d_in[5];
  const float* b2     = (const float*)d_in[6];
  const float* county = (const float*)d_in[7];
  const float* W_ih   = (const float*)d_in[8];
  const float* W_hh   = (const float*)d_in[9];
  const float* b_ih   = (const float*)d_in[10];
  const float* b_hh   = (const float*)d_in[11];
  const float* Wm1    = (const float*)d_in[12];
  const float* bm1    = (const float*)d_in[13];
  const float* Wm2    = (const float*)d_in[14];
  const float* bm2    = (const float*)d_in[15];

  const int E  = in_sizes[2];
  const int N  = in_sizes[7] / DD;
  const int T  = (int)(in_sizes[0] / ((long)N * FEAT));
  const long TN = (long)T * N;

  const int* srcA = ei;        // edge_index[0]
  const int* dstA = ei + E;    // edge_index[1]

  // workspace layout (256B aligned slices)
  size_t o = 0;
  auto take = [&](size_t bytes) {
    void* r = (char*)d_ws + o;
    o += (bytes + 255) & ~(size_t)255;
    return r;
  };
  float* dis   = (float*)take((size_t)N * 4);
  float* normw = (float*)take((size_t)(E + N) * 4);
  int*   icnt  = (int*)  take((size_t)N * 4);
  int*   iptr  = (int*)  take((size_t)(N + 1) * 4);
  int*   icur  = (int*)  take((size_t)N * 4);
  int*   ieid  = (int*)  take((size_t)(E + N) * 4);
  float* hfin  = (float*)take((size_t)N * HHID * 4);
  float* bufA  = (float*)take((size_t)TN * DD * 4);
  float* bufB  = (float*)take((size_t)TN * DD * 4);
  (void)ws_size; (void)n_in; (void)out_size;

  // --- graph normalization + deterministic CSR ---
  (void)hipMemsetAsync(icnt, 0, (size_t)N * 4, stream);
  int en = E + N;
  k_count  <<<(en + 255) / 256, 256, 0, stream>>>(dstA, icnt, E, N);
  k_scan   <<<1, 1, 0, stream>>>(icnt, iptr, icur, N);
  k_scatter<<<(en + 255) / 256, 256, 0, stream>>>(dstA, icur, ieid, E, N);
  k_sort_deg<<<(N + 255) / 256, 256, 0, stream>>>(ieid, iptr, ew, dis, E, N);
  k_norm   <<<(en + 255) / 256, 256, 0, stream>>>(srcA, dstA, ew, dis, normw, E, N);

  // --- GCN layer 1: transform -> aggregate(+b1, relu) ---
  long n_hw1 = TN * 64;
  k_hw1<<<(unsigned)((n_hw1 + 255) / 256), 256, 0, stream>>>(x, W1, bufA, TN);
  dim3 gagg((N + 3) / 4, T);
  k_agg<<<gagg, 256, 0, stream>>>(bufA, iptr, ieid, srcA, normw, b1,
                                  nullptr, 0, 1, bufB, N, E);

  // --- GCN layer 2 (WMMA): transform -> aggregate(+b2+county) -> embeds ---
  int ntiles = (int)(TN / 16);
  k_hw2<<<(ntiles + 7) / 8, 256, 0, stream>>>(bufB, W2, bufA, ntiles);
  k_agg<<<gagg, 256, 0, stream>>>(bufA, iptr, ieid, srcA, normw, b2,
                                  county, 1, 0, bufB, N, E);

  // --- persistent WMMA LSTM over all T steps (async emb double-buffer) ---
  size_t lstm_lds = (size_t)(512 * 132 + 16 * 132 + 2 * 16 * 68) * 4;  // 287488 B
  k_lstm<<<(N + 15) / 16, 256, lstm_lds, stream>>>(bufB, W_ih, W_hh, b_ih, b_hh,
                                                   hfin, T, N);

  // --- head MLP ---
  k_head<<<N, 64, 0, stream>>>(hfin, Wm1, bm1, Wm2, bm2, (float*)d_out, N);
}